// HeatmapMSELoss_72739566125205
// MI455X (gfx1250) — compile-verified
//
#include <hip/hip_runtime.h>
#include <math.h>

// ---- problem constants (from reference) ----
#define B_ 32
#define V_ 4
#define J_ 17
#define H_ 128
#define W_ 128
static constexpr int   SLICES      = B_ * V_ * J_;          // 2176
static constexpr int   SLICE_ELEMS = H_ * W_;               // 16384 (64 KB f32)
static constexpr float INV_N       = 1.0f / (float)((long long)B_ * V_ * J_ * H_ * W_);

typedef float f4  __attribute__((ext_vector_type(4)));
typedef float v2f __attribute__((ext_vector_type(2)));
typedef float v8f __attribute__((ext_vector_type(8)));

// ---------------------------------------------------------------------------
// Wave32 allreduce using V_WMMA_F32_16X16X4_F32.
// A (16x4): VGPR0 = partial (lanes 0-15 -> A[m,0], lanes 16-31 -> A[m,2]),
//           VGPR1 = 0 (K=1,3). B (4x16) = all ones.
// => D[m,n] = p[m] + p[m+16]  (independent of n).
// Sum of the 8 D VGPRs: lanes 0-15 hold sum over m=0..7, lanes 16-31 hold
// sum over m=8..15; one xor-16 swap+add gives the full 32-lane sum in every
// lane.  EXEC is all-ones at every call site (no divergence there).
// ---------------------------------------------------------------------------
__device__ __forceinline__ float wave_allreduce_wmma(float p) {
    v2f a; a[0] = p;    a[1] = 0.0f;
    v2f b; b[0] = 1.0f; b[1] = 1.0f;
    v8f c = {};
    v8f d = __builtin_amdgcn_wmma_f32_16x16x4_f32(
        /*neg_a=*/false, a, /*neg_b=*/false, b,
        /*c_mod=*/(short)0, c, /*reuse_a=*/false, /*reuse_b=*/false);
    float s = d[0] + d[1] + d[2] + d[3] + d[4] + d[5] + d[6] + d[7];
    s += __shfl_xor(s, 16, 32);
    return s;
}

// Deterministic block reduction: WMMA wave allreduce -> LDS -> sequential sum
// by thread 0 (fixed order => bitwise deterministic across replays).
__device__ __forceinline__ float block_reduce_256(float p, float* sWave) {
    float w = wave_allreduce_wmma(p);
    const int lane = threadIdx.x & 31;
    const int wid  = threadIdx.x >> 5;
    if (lane == 0) sWave[wid] = w;
    __syncthreads();
    float total = 0.0f;
    if (threadIdx.x == 0) {
        #pragma unroll
        for (int i = 0; i < 8; ++i) total += sWave[i];
    }
    return total; // valid in thread 0 only
}

// ---------------------------------------------------------------------------
// One block per (b,v,j) slice: project the joint, build separable Gaussian
// tables ex[128]/ey[128] in LDS, then stream the 64 KB slice with NT b128
// loads accumulating (pred - ey*ex)^2.
// ---------------------------------------------------------------------------
__global__ void __launch_bounds__(256)
heatmap_mse_partial(const float* __restrict__ pred,   // [B,V,J,H,W]
                    const float* __restrict__ proj,   // [B,V,3,4]
                    const float* __restrict__ j3d,    // [B,J,3]
                    float* __restrict__ partials)     // [SLICES]
{
    __shared__ __align__(16) float sEx[W_];
    __shared__ __align__(16) float sEy[H_];
    __shared__ float sWave[8];

    const int s   = blockIdx.x;              // slice id = ((b*V)+v)*J + j
    const int b   = s / (V_ * J_);
    const int rem = s - b * (V_ * J_);
    const int v   = rem / J_;
    const int j   = rem - v * J_;

    // ---- uniform per-block projection (scalar-friendly loads) ----
    const float* P = proj + (size_t)(b * V_ + v) * 12;
    const float* G = j3d  + (size_t)(b * J_ + j) * 3;
    const float X = G[0], Y = G[1], Z = G[2];
    const float px = P[0]*X + P[1]*Y + P[2]*Z  + P[3];
    const float py = P[4]*X + P[5]*Y + P[6]*Z  + P[7];
    const float pz = P[8]*X + P[9]*Y + P[10]*Z + P[11];
    const float inv = 1.0f / pz;             // row 2 ensures pz ~ 1.1..1.4
    const float cx = px * inv;
    const float cy = py * inv;

    // ---- separable Gaussian tables: 256 exps per slice instead of 16384 ----
    const int t = threadIdx.x;
    if (t < W_) {
        float d = (float)t - cx;
        sEx[t] = __expf(-0.5f * d * d);      // inv_two_sigma2 = 0.5 (SIGMA=1)
    } else {
        int h = t - W_;
        float d = (float)h - cy;
        sEy[h] = __expf(-0.5f * d * d);
    }
    __syncthreads();

    // ---- stream the slice: 4096 float4 / 256 threads = 16 b128 loads each.
    // i = t + it*256: each wave covers exactly one row per iteration
    // (32 float4 == 128 floats), so ey is wave-uniform (broadcast DS read),
    // and ex is read as one b128 per lane (lane L -> sEx[4L..4L+3]).
    const f4* p4 = (const f4*)(pred + (size_t)s * SLICE_ELEMS);
    const f4* e4 = (const f4*)sEx;
    float acc = 0.0f;
    #pragma unroll
    for (int it = 0; it < 16; ++it) {
        const int i  = t + it * 256;                       // float4 index
        const f4 pv  = __builtin_nontemporal_load(&p4[i]); // global_load_b128 NT
        const float ey = sEy[i >> 5];
        const f4 ex    = e4[i & 31];
        float d0 = fmaf(-ey, ex[0], pv[0]);
        float d1 = fmaf(-ey, ex[1], pv[1]);
        float d2 = fmaf(-ey, ex[2], pv[2]);
        float d3 = fmaf(-ey, ex[3], pv[3]);
        acc = fmaf(d0, d0, acc);
        acc = fmaf(d1, d1, acc);
        acc = fmaf(d2, d2, acc);
        acc = fmaf(d3, d3, acc);
    }

    const float total = block_reduce_256(acc, sWave);
    if (t == 0) partials[s] = total;
}

// ---------------------------------------------------------------------------
// Single-block finalize: fixed-order reduction of 2176 partials, scale by 1/N.
// Fully overwrites d_out[0] (harness poisons it), no atomics => deterministic.
// ---------------------------------------------------------------------------
__global__ void __launch_bounds__(256)
heatmap_mse_finalize(const float* __restrict__ partials, float* __restrict__ out)
{
    __shared__ float sWave[8];
    float acc = 0.0f;
    for (int i = threadIdx.x; i < SLICES; i += 256) acc += partials[i];
    const float total = block_reduce_256(acc, sWave);
    if (threadIdx.x == 0) out[0] = total * INV_N;
}

extern "C" void kernel_launch(void* const* d_in, const int* in_sizes, int n_in,
                              void* d_out, int out_size, void* d_ws, size_t ws_size,
                              hipStream_t stream) {
    (void)in_sizes; (void)n_in; (void)out_size; (void)ws_size;
    const float* pred = (const float*)d_in[0];   // heatmaps_pred  [B,V,J,H,W]
    const float* proj = (const float*)d_in[1];   // proj_mats      [B,V,3,4]
    const float* j3d  = (const float*)d_in[2];   // joints_3d_gt   [B,J,3]
    // d_in[3] (joints_3d_valid) is unused by the reference math.

    float* partials = (float*)d_ws;              // needs SLICES*4 = 8.7 KB

    heatmap_mse_partial<<<SLICES, 256, 0, stream>>>(pred, proj, j3d, partials);
    heatmap_mse_finalize<<<1, 256, 0, stream>>>(partials, (float*)d_out);
}